// SSMBlock_1975684956469
// MI455X (gfx1250) — compile-verified
//
#include <hip/hip_runtime.h>
#include <hip/hip_bf16.h>

typedef __attribute__((ext_vector_type(2))) float v2f;
typedef __attribute__((ext_vector_type(8))) float v8f;

#define DIM     1024
#define DSTATE  16
#define BATCH   4
#define SEQ     8192
#define CHUNK   128
#define NCHUNK  (SEQ / CHUNK)     // 64
#define SD_STRIDE 1032            // padded D row stride in LDS (floats)

// ---------------------------------------------------------------------------
// Phase 1: per-chunk  x_state = x @ A  (WMMA fp32), in-chunk cumsum,
//          write cumsummed states to ws1, chunk totals to ws2.
// grid = BATCH*NCHUNK blocks of 128 threads (4 waves).
// ---------------------------------------------------------------------------
__global__ void __launch_bounds__(128)
ssm_phase1(const float* __restrict__ x, const float* __restrict__ A,
           float* __restrict__ ws1, float* __restrict__ ws2)
{
    __shared__ alignas(16) float sA[DIM * DSTATE];   // 64 KB, A staged
    __shared__ alignas(16) float sS[CHUNK * DSTATE]; // 8 KB, chunk states
    __shared__ alignas(16) float sTot[8 * DSTATE];

    const int blk   = blockIdx.x;
    const int b     = blk >> 6;        // / NCHUNK
    const int chunk = blk & (NCHUNK - 1);
    const int n0    = chunk * CHUNK;
    const int tid   = threadIdx.x;

    // stage A into LDS (coalesced float4)
    {
        const float4* src = (const float4*)A;
        float4*       dst = (float4*)sA;
        for (int i = tid; i < DIM * DSTATE / 4; i += 128) dst[i] = src[i];
    }
    __syncthreads();

    const int wave = tid >> 5;
    const int lane = tid & 31;
    const int lrow = lane & 15;
    const int hi   = lane >> 4;        // lane half: 0 or 1

    // each wave computes two 16-row tiles: t = wave*2 + tt
    for (int tt = 0; tt < 2; ++tt) {
        const int t = wave * 2 + tt;                       // 0..7
        const float* xrow = x + ((size_t)b * SEQ + n0 + t * 16 + lrow) * DIM;
        v8f acc = {};
        #pragma unroll 4
        for (int kk = 0; kk < DIM / 4; ++kk) {
            // A-operand: 16x4 tile of x. VGPR0=K(0|2), VGPR1=K(1|3)
            const int k0 = kk * 4 + hi * 2;
            v2f a = *(const v2f*)(xrow + k0);
            // B-operand: 4x16 tile of A. VGPR0=row K(0|2), VGPR1=row K(1|3)
            v2f bb;
            bb.x = sA[(k0    ) * DSTATE + lrow];
            bb.y = sA[(k0 + 1) * DSTATE + lrow];
            acc = __builtin_amdgcn_wmma_f32_16x16x4_f32(
                false, a, false, bb, (short)0, acc, false, false);
        }
        // C/D layout: VGPR j -> row (j | j+8), col = lane&15
        const int rbase = t * 16 + hi * 8;
        #pragma unroll
        for (int j = 0; j < 8; ++j)
            sS[(rbase + j) * DSTATE + lrow] = acc[j];
    }
    __syncthreads();

    // hierarchical in-chunk cumsum: thread = (state s, row-group g)
    {
        const int s = tid & 15;
        const int g = tid >> 4;        // 0..7, 16 rows each
        float run = 0.f;
        #pragma unroll
        for (int r = 0; r < 16; ++r) {
            run += sS[(g * 16 + r) * DSTATE + s];
            sS[(g * 16 + r) * DSTATE + s] = run;
        }
        sTot[g * DSTATE + s] = run;
        __syncthreads();
        float off = 0.f;
        for (int gg = 0; gg < g; ++gg) off += sTot[gg * DSTATE + s];
        #pragma unroll
        for (int r = 0; r < 16; ++r)
            sS[(g * 16 + r) * DSTATE + s] += off;
        if (g == 7)   // chunk total = offset of last group + its local total
            ws2[((size_t)b * NCHUNK + chunk) * DSTATE + s] = off + run;
    }
    __syncthreads();

    // write cumsummed states to ws1 (contiguous 8 KB, coalesced)
    {
        float4*       dst = (float4*)(ws1 + ((size_t)b * SEQ + n0) * DSTATE);
        const float4* src = (const float4*)sS;
        for (int i = tid; i < CHUNK * DSTATE / 4; i += 128) dst[i] = src[i];
    }
}

// ---------------------------------------------------------------------------
// Phase 2: out = (ws1 + prefix_offset(ws2)) @ D  via WMMA fp32.
// grid = BATCH*NCHUNK blocks of 256 threads (8 waves, one 16-row band each).
// ---------------------------------------------------------------------------
__global__ void __launch_bounds__(256)
ssm_phase2(const float* __restrict__ ws1, const float* __restrict__ ws2,
           const float* __restrict__ Dm, float* __restrict__ out)
{
    __shared__ alignas(16) float sD[DSTATE * SD_STRIDE]; // padded: no bank alias
    __shared__ alignas(16) float sS[CHUNK * DSTATE];
    __shared__ float sOff[DSTATE];

    const int blk   = blockIdx.x;
    const int b     = blk >> 6;
    const int chunk = blk & (NCHUNK - 1);
    const int n0    = chunk * CHUNK;
    const int tid   = threadIdx.x;

    // stage D into LDS with padded row stride
    {
        const float4* src = (const float4*)Dm;
        for (int i4 = tid; i4 < DSTATE * DIM / 4; i4 += 256) {
            const int row  = i4 >> 8;         // /(DIM/4)
            const int col4 = i4 & 255;
            ((float4*)(sD + row * SD_STRIDE))[col4] = src[i4];
        }
    }
    // stage chunk states
    {
        const float4* src = (const float4*)(ws1 + ((size_t)b * SEQ + n0) * DSTATE);
        float4*       dst = (float4*)sS;
        for (int i = tid; i < CHUNK * DSTATE / 4; i += 256) dst[i] = src[i];
    }
    // per-state prefix offset over preceding chunks
    if (tid < DSTATE) {
        float off = 0.f;
        for (int c = 0; c < chunk; ++c)
            off += ws2[((size_t)b * NCHUNK + c) * DSTATE + tid];
        sOff[tid] = off;
    }
    __syncthreads();
    if (tid < 128) {
        const int s = tid & 15;
        const int g = tid >> 4;
        const float off = sOff[s];
        #pragma unroll
        for (int r = 0; r < 16; ++r) sS[(g * 16 + r) * DSTATE + s] += off;
    }
    __syncthreads();

    const int wave = tid >> 5;   // 0..7: 16-row band of the chunk
    const int lane = tid & 31;
    const int lrow = lane & 15;
    const int hi   = lane >> 4;

    // resident A-operands: state tile rows, K split into 4 chunks of 4
    const float* srow = sS + (wave * 16 + lrow) * DSTATE;
    v2f a0 = *(const v2f*)(srow + 0  + hi * 2);
    v2f a1 = *(const v2f*)(srow + 4  + hi * 2);
    v2f a2 = *(const v2f*)(srow + 8  + hi * 2);
    v2f a3 = *(const v2f*)(srow + 12 + hi * 2);

    float* obase = out + ((size_t)b * SEQ + n0 + wave * 16 + hi * 8) * DIM;
    for (int j = 0; j < DIM / 16; ++j) {
        const int ncol = j * 16 + lrow;
        v2f bb;
        v8f acc = {};
        bb.x = sD[(0  + hi * 2) * SD_STRIDE + ncol];
        bb.y = sD[(1  + hi * 2) * SD_STRIDE + ncol];
        acc = __builtin_amdgcn_wmma_f32_16x16x4_f32(false, a0, false, bb, (short)0, acc, false, false);
        bb.x = sD[(4  + hi * 2) * SD_STRIDE + ncol];
        bb.y = sD[(5  + hi * 2) * SD_STRIDE + ncol];
        acc = __builtin_amdgcn_wmma_f32_16x16x4_f32(false, a1, false, bb, (short)0, acc, false, false);
        bb.x = sD[(8  + hi * 2) * SD_STRIDE + ncol];
        bb.y = sD[(9  + hi * 2) * SD_STRIDE + ncol];
        acc = __builtin_amdgcn_wmma_f32_16x16x4_f32(false, a2, false, bb, (short)0, acc, false, false);
        bb.x = sD[(12 + hi * 2) * SD_STRIDE + ncol];
        bb.y = sD[(13 + hi * 2) * SD_STRIDE + ncol];
        acc = __builtin_amdgcn_wmma_f32_16x16x4_f32(false, a3, false, bb, (short)0, acc, false, false);
        #pragma unroll
        for (int r = 0; r < 8; ++r)
            obase[(size_t)r * DIM + ncol] = acc[r];
    }
}

// ---------------------------------------------------------------------------
extern "C" void kernel_launch(void* const* d_in, const int* in_sizes, int n_in,
                              void* d_out, int out_size, void* d_ws, size_t ws_size,
                              hipStream_t stream)
{
    (void)in_sizes; (void)n_in; (void)out_size; (void)ws_size;
    const float* x  = (const float*)d_in[0];
    const float* A  = (const float*)d_in[1];
    const float* Dm = (const float*)d_in[2];
    float* out = (float*)d_out;

    float* ws1 = (float*)d_ws;                               // [B, SEQ, 16]
    float* ws2 = ws1 + (size_t)BATCH * SEQ * DSTATE;         // [B, NCHUNK, 16]

    dim3 grid(BATCH * NCHUNK);
    ssm_phase1<<<grid, 128, 0, stream>>>(x, A, ws1, ws2);
    ssm_phase2<<<grid, 256, 0, stream>>>(ws1, ws2, Dm, out);
}